// Damped_electrostatics_NoShift_48498770706886
// MI455X (gfx1250) — compile-verified
//
#include <hip/hip_runtime.h>
#include <hip/hip_bf16.h>
#include <stdint.h>

// Damped electrostatics (no shift), MI455X / gfx1250.
//
// Roofline: ~28 B/edge streaming HBM (d, vec, idx, out) + ~68 B/edge gathered
// from a 5.2 MB L2-resident atom set; ~120 fp32 FLOPs/edge. Pure HBM-bound
// (~90 MB / 23.3 TB/s ~= 4 us floor). No GEMM structure -> WMMA inapplicable.
// CDNA5-specific path used instead: GLOBAL_LOAD_ASYNC_TO_LDS double-buffered
// tile staging (ASYNCcnt / s_wait_asynccnt), lane-local so no barriers needed,
// plus non-temporal output stores.

#define KEHALF_F 7.199822675975274f
#define CUTOFF_F 12.0f
#define INV_SR_F 0.5f // 1 / CUTOFF_SR

#define TILE      256
#define BUF_BYTES 6144           // 256*4 (dist) + 256*12 (vec) + 256*4 (idxu) + 256*4 (idxv)
#define OFF_DIST  0
#define OFF_VEC   1024
#define OFF_IDXU  4096
#define OFF_IDXV  5120

// ---- CDNA5 async global->LDS (GVS mode: SGPR base + 32-bit VGPR offset) ----
__device__ __forceinline__ void async_b32(uint32_t lds_addr, uint32_t gbyte_off,
                                          uint64_t sbase) {
    asm volatile("global_load_async_to_lds_b32 %0, %1, %2"
                 :: "v"(lds_addr), "v"(gbyte_off), "s"(sbase) : "memory");
}
__device__ __forceinline__ void wait_async_le6() {
    asm volatile("s_wait_asynccnt 0x6" ::: "memory");
}
__device__ __forceinline__ void wait_async_0() {
    asm volatile("s_wait_asynccnt 0x0" ::: "memory");
}

// ---- per-edge physics ----
__device__ __forceinline__ float edge_energy(
    float d, float vx, float vy, float vz, int u, int v,
    const float* __restrict__ q, const float* __restrict__ dip,
    const float* __restrict__ quad)
{
    const float qu = q[u];
    const float qv = q[v];
    const float mux_u = dip[3 * u + 0], muy_u = dip[3 * u + 1], muz_u = dip[3 * u + 2];
    const float mux_v = dip[3 * v + 0], muy_v = dip[3 * v + 1], muz_v = dip[3 * v + 2];
    const float* __restrict__ Q = quad + 9 * v;
    const float Q00 = Q[0], Q01 = Q[1], Q02 = Q[2];
    const float Q10 = Q[3], Q11 = Q[4], Q12 = Q[5];
    const float Q20 = Q[6], Q21 = Q[7], Q22 = Q[8];

    // quintic switch
    float x = fminf(fmaxf(d * INV_SR_F, 0.0f), 1.0f);
    const float x2 = x * x;
    const float sw = 1.0f - x2 * x * fmaf(x, fmaf(6.0f, x, -15.0f), 10.0f);

    const float inv_d    = 1.0f / d;
    const float d_damped = sqrtf(fmaf(d, d, 1.0f));
    const float chi  = sw / d_damped + (1.0f - sw) * inv_d;
    const float chi2 = chi * chi;
    const float chi3 = chi2 * chi;

    // charge-charge
    float Eelec = qu * qv * chi;

    // dipole terms
    const float dot_uv = fmaf(vx, mux_v, fmaf(vy, muy_v, vz * muz_v)) * inv_d;
    const float dot_vu = fmaf(vx, mux_u, fmaf(vy, muy_u, vz * muz_u)) * inv_d;
    const float mumu   = fmaf(mux_u, mux_v, fmaf(muy_u, muy_v, muz_u * muz_v));
    Eelec = fmaf(2.0f * qu * dot_uv, chi2, Eelec);
    Eelec = fmaf(fmaf(-3.0f * dot_uv, dot_vu, mumu), chi3, Eelec);

    // quadrupole: sum_ij [(v_i v_j - |v|^2/3 delta_ij)/d^2] Q_ij
    //           = (v^T Q v - |v|^2/3 tr(Q)) / d^2
    const float vQv =
          vx * fmaf(Q00, vx, fmaf(Q01, vy, Q02 * vz))
        + vy * fmaf(Q10, vx, fmaf(Q11, vy, Q12 * vz))
        + vz * fmaf(Q20, vx, fmaf(Q21, vy, Q22 * vz));
    const float v2  = fmaf(vx, vx, fmaf(vy, vy, vz * vz));
    const float trQ = Q00 + Q11 + Q22;
    const float sum_uv = (vQv - v2 * (1.0f / 3.0f) * trQ) * (inv_d * inv_d);
    Eelec = fmaf(qu * sum_uv, chi3, Eelec);

    Eelec *= KEHALF_F;
    return (d <= CUTOFF_F) ? Eelec : 0.0f;
}

__global__ __launch_bounds__(256) void
Damped_electrostatics_NoShift_48498770706886_kernel(
    const float* __restrict__ dist,      // [E]
    const float* __restrict__ vec,       // [E,3]
    const float* __restrict__ q,         // [N]
    const float* __restrict__ dip,       // [N,3]
    const float* __restrict__ quad,      // [N,3,3]
    const int*   __restrict__ idx_u,     // [E]
    const int*   __restrict__ idx_v,     // [E]
    float*       __restrict__ out,       // [E]
    int E)
{
    __shared__ __align__(16) unsigned char smem[2][BUF_BYTES];

    const int lane     = (int)threadIdx.x;
    const int numTiles = E / TILE;                 // full tiles via async pipeline
    const int tStride  = (int)gridDim.x;

    const uint64_t bDist = (uint64_t)(uintptr_t)dist;
    const uint64_t bVec  = (uint64_t)(uintptr_t)vec;
    const uint64_t bIdxU = (uint64_t)(uintptr_t)idx_u;
    const uint64_t bIdxV = (uint64_t)(uintptr_t)idx_v;

    // Low 32 bits of a generic LDS pointer == wave-relative LDS byte address
    // (ISA aperture mapping).
    const uint32_t ldsBase0 = (uint32_t)(uintptr_t)&smem[0][0];
    const uint32_t ldsBase1 = (uint32_t)(uintptr_t)&smem[1][0];

    // Stage one 256-edge tile into LDS buffer `buf`: 6 async ops per wave.
    auto stage = [&](int tile, uint32_t lbase) {
        const uint32_t e = (uint32_t)tile * TILE + (uint32_t)lane;
        async_b32(lbase + OFF_DIST + 4u * (uint32_t)lane, 4u * e,       bDist);
        const uint32_t lvec = lbase + OFF_VEC + 12u * (uint32_t)lane;
        const uint32_t gvec = 12u * e;
        async_b32(lvec + 0u, gvec + 0u, bVec);
        async_b32(lvec + 4u, gvec + 4u, bVec);
        async_b32(lvec + 8u, gvec + 8u, bVec);
        async_b32(lbase + OFF_IDXU + 4u * (uint32_t)lane, 4u * e, bIdxU);
        async_b32(lbase + OFF_IDXV + 4u * (uint32_t)lane, 4u * e, bIdxV);
    };

    int t   = (int)blockIdx.x;
    int buf = 0;
    if (t < numTiles) stage(t, ldsBase0);

    for (; t < numTiles; t += tStride, buf ^= 1) {
        const int tn = t + tStride;
        if (tn < numTiles) {
            stage(tn, buf ? ldsBase0 : ldsBase1);  // fill the other buffer
            wait_async_le6();                      // oldest 6 ops (this tile) done
        } else {
            wait_async_0();                        // drain: this tile done
        }

        const float* sdist = (const float*)&smem[buf][OFF_DIST];
        const float* svec  = (const float*)&smem[buf][OFF_VEC];
        const int*   siu   = (const int*)  &smem[buf][OFF_IDXU];
        const int*   siv   = (const int*)  &smem[buf][OFF_IDXV];

        const float d  = sdist[lane];
        const float vx = svec[3 * lane + 0];
        const float vy = svec[3 * lane + 1];
        const float vz = svec[3 * lane + 2];
        const int   u  = siu[lane];
        const int   v  = siv[lane];

        const float r = edge_energy(d, vx, vy, vz, u, v, q, dip, quad);
        __builtin_nontemporal_store(r, &out[t * TILE + lane]);
    }

    // Tail (E % TILE edges): direct-load path.
    const int tailStart = numTiles * TILE;
    for (int e = tailStart + (int)(blockIdx.x * blockDim.x + threadIdx.x); e < E;
         e += (int)(gridDim.x * blockDim.x)) {
        const float d  = dist[e];
        const float vx = vec[3 * e + 0];
        const float vy = vec[3 * e + 1];
        const float vz = vec[3 * e + 2];
        const float r  = edge_energy(d, vx, vy, vz, idx_u[e], idx_v[e], q, dip, quad);
        __builtin_nontemporal_store(r, &out[e]);
    }
}

extern "C" void kernel_launch(void* const* d_in, const int* in_sizes, int n_in,
                              void* d_out, int out_size, void* d_ws, size_t ws_size,
                              hipStream_t stream) {
    const float* dist = (const float*)d_in[0];
    const float* vec  = (const float*)d_in[1];
    const float* q    = (const float*)d_in[2];
    const float* dip  = (const float*)d_in[3];
    const float* quad = (const float*)d_in[4];
    const int*   iu   = (const int*)d_in[5];
    const int*   iv   = (const int*)d_in[6];
    float*       out  = (float*)d_out;

    const int E = in_sizes[0];
    const int threads = 256;                 // 8 wave32 waves per block
    int blocks = (E + threads - 1) / threads;
    if (blocks > 4096) blocks = 4096;        // grid-stride over 256-edge tiles
    if (blocks < 1) blocks = 1;

    hipLaunchKernelGGL(Damped_electrostatics_NoShift_48498770706886_kernel,
                       dim3(blocks), dim3(threads), 0, stream,
                       dist, vec, q, dip, quad, iu, iv, out, E);
}